// LightweightConv1d_69741678952739
// MI455X (gfx1250) — compile-verified
//
#include <hip/hip_runtime.h>
#include <cstdint>
#include <type_traits>

// LightweightConv1d: B=8, T=4096, C=1024, H=16 heads, K=7, pad=3, softmax'd weights.
// Memory-bound (AI ~1.75 F/B) -> optimize the data path, not the math:
//   stage 1: 16-thread kernel softmaxes the taps once into d_ws
//   stage 2: global_load_async_to_lds_b128 (ASYNCcnt DMA) -> LDS tile,
//            s_wait_asynccnt 0 + barrier, rolling 7-tap register window,
//            nontemporal b128 stores. Interior t-chunks take a check-free path.

typedef __attribute__((ext_vector_type(4))) float v4f;

#define B_DIM   8
#define T_DIM   4096
#define C_DIM   1024
#define H_DIM   16
#define K_DIM   7
#define HALO    3

#define CC      64                  // channels per block tile
#define TT      128                 // timesteps per block tile
#define ROWS    (TT + 2 * HALO)     // 134 input rows incl. halo
#define COLS    (CC / 4)            // 16 v4f per row
#define PADCOLS (COLS + 1)          // +1 v4f pad to stagger LDS banks
#define NLOAD   (ROWS * COLS)       // 2144 v4f tiles to DMA
#define NROUNDS ((NLOAD + 255) / 256)
#define NTHREADS 256

// ---- stage 1: softmax the (16,7) taps once ----
__global__ void softmax_weights_kernel(const float* __restrict__ w,
                                       float* __restrict__ wsoft)
{
    const int h = threadIdx.x;
    if (h < H_DIM) {
        float v[K_DIM];
        float m = -1e30f;
        #pragma unroll
        for (int k = 0; k < K_DIM; ++k) { v[k] = w[h * K_DIM + k]; m = fmaxf(m, v[k]); }
        float s = 0.f;
        #pragma unroll
        for (int k = 0; k < K_DIM; ++k) { v[k] = expf(v[k] - m); s += v[k]; }
        const float inv = 1.f / s;
        #pragma unroll
        for (int k = 0; k < K_DIM; ++k) wsoft[h * K_DIM + k] = v[k] * inv;
    }
}

// ---- stage 2: streaming depthwise conv ----
__global__ __launch_bounds__(NTHREADS)
void lwconv_async_kernel(const float* __restrict__ x,
                         const float* __restrict__ wsoft,
                         float* __restrict__ y)
{
    __shared__ v4f smem[ROWS][PADCOLS];

    const int tid    = threadIdx.x;
    const int bz     = blockIdx.x;
    const int tchunk = bz & 31;          // T/TT = 32
    const int cchunk = (bz >> 5) & 15;   // C/CC = 16
    const int b      = bz >> 9;          // B = 8
    const int t0     = tchunk * TT;
    const int cbase  = cchunk * CC;

    const int c4 = tid & 15;             // which v4f column of the tile
    const int tg = tid >> 4;             // which 8-row stripe of outputs

    // precomputed taps: 4 consecutive heads = 28 contiguous floats
    const int    h0 = (c4 * 4) & (H_DIM - 1);
    const float* wp = wsoft + h0 * K_DIM;
    v4f ws[K_DIM];
    #pragma unroll
    for (int k = 0; k < K_DIM; ++k)
        ws[k] = (v4f){wp[k], wp[K_DIM + k], wp[2 * K_DIM + k], wp[3 * K_DIM + k]};

    const float*   gbase   = x + ((size_t)b * T_DIM + (size_t)(t0 - HALO)) * C_DIM + cbase;
    const unsigned ldsbase = (unsigned)(uintptr_t)(&smem[0][0]);
    const int      trow0   = tg * 8;
    float* outbase = y + ((size_t)b * T_DIM + (size_t)(t0 + trow0)) * C_DIM
                       + cbase + c4 * 4;

    auto run = [&](auto chk_tag) {
        constexpr bool CHK = decltype(chk_tag)::value;

        // async DMA: (ROWS x CC) fp32 tile, global -> LDS, b128 per lane
        #pragma unroll
        for (int r = 0; r < NROUNDS; ++r) {
            const int l   = tid + r * NTHREADS;
            const int row = l >> 4;          // / COLS
            const int col = l & 15;          // % COLS
            bool go = (l < NLOAD);
            if constexpr (CHK) {
                const int t = t0 - HALO + row;
                go = go && (t >= 0) && (t < T_DIM);
            }
            if (go) {
                const unsigned voff  = (unsigned)(row * (C_DIM * 4) + col * 16);
                const unsigned laddr = ldsbase + (unsigned)((row * PADCOLS + col) * 16);
                asm volatile("global_load_async_to_lds_b128 %0, %1, %2"
                             :: "v"(laddr), "v"(voff), "s"(gbase)
                             : "memory");
            }
        }
        asm volatile("s_wait_asynccnt 0" ::: "memory");  // my wave's DMAs landed
        __syncthreads();                                 // everyone's landed (uniform branch)

        auto ldrow = [&](int r) -> v4f {
            if constexpr (CHK) {
                const int t = t0 - HALO + r;
                if (t < 0 || t >= T_DIM) return (v4f){0.f, 0.f, 0.f, 0.f};
            }
            return smem[r][c4];
        };

        v4f win[K_DIM];
        #pragma unroll
        for (int k = 0; k < K_DIM - 1; ++k) win[k] = ldrow(trow0 + k);

        #pragma unroll
        for (int j = 0; j < 8; ++j) {
            win[K_DIM - 1] = ldrow(trow0 + j + K_DIM - 1);

            v4f acc = (v4f){0.f, 0.f, 0.f, 0.f};
            #pragma unroll
            for (int k = 0; k < K_DIM; ++k) acc += ws[k] * win[k];

            __builtin_nontemporal_store(acc,
                reinterpret_cast<v4f*>(outbase + (size_t)j * C_DIM));

            #pragma unroll
            for (int k = 0; k < K_DIM - 1; ++k) win[k] = win[k + 1];
        }
    };

    // interior chunks (30 of 32) need no boundary clamps anywhere
    if (t0 >= HALO && t0 + TT + HALO <= T_DIM) {
        run(std::integral_constant<bool, false>{});
    } else {
        run(std::integral_constant<bool, true>{});
    }
}

extern "C" void kernel_launch(void* const* d_in, const int* in_sizes, int n_in,
                              void* d_out, int out_size, void* d_ws, size_t ws_size,
                              hipStream_t stream)
{
    (void)in_sizes; (void)n_in; (void)out_size; (void)ws_size;
    const float* x = (const float*)d_in[0];   // (8, 4096, 1024) fp32
    const float* w = (const float*)d_in[1];   // (16, 1, 7) fp32
    float*       y = (float*)d_out;           // (8, 4096, 1024) fp32
    float*   wsoft = (float*)d_ws;            // 112 floats of scratch

    hipLaunchKernelGGL(softmax_weights_kernel, dim3(1), dim3(H_DIM), 0, stream, w, wsoft);

    const int nblocks = B_DIM * (C_DIM / CC) * (T_DIM / TT);  // 4096
    hipLaunchKernelGGL(lwconv_async_kernel,
                       dim3(nblocks), dim3(NTHREADS), 0, stream,
                       x, wsoft, y);
}